// BaseKaolinRenderer_60206851555807
// MI455X (gfx1250) — compile-verified
//
#include <hip/hip_runtime.h>
#include <hip/hip_bf16.h>
#include <math.h>

#define BATCH  4
#define NV     4000
#define NF     6000
#define IMH    64
#define IMW    64
#define NPIX   (IMH*IMW)
#define SIGMA  0.01f
#define NGROUP  (NF/4)        // 1500 real groups of 4 faces
#define NGROUPP 1504          // padded to multiple of CHUNK
#define NFP     (NGROUPP*4)   // 6016 faces incl. sentinels
#define CHUNK   16            // groups per LDS chunk (16*256B = 4KB)
#define NCHUNK  (NGROUPP/CHUNK) // 94

typedef float v2f __attribute__((ext_vector_type(2)));
typedef float v8f __attribute__((ext_vector_type(8)));

// workspace layout (float offsets)
#define WS_VSCAM  0                                // B*V*3 camera-space verts
#define WS_VSIMG  (WS_VSCAM + BATCH*NV*3)          // B*V*2 image-space verts
#define WS_VNORM  (WS_VSIMG + BATCH*NV*2)          // B*V*3 vertex normals
#define WS_COEF   (WS_VNORM + BATCH*NV*3)          // B*NGROUPP*64 WMMA-A coeffs (lane-order)
#define WS_FATTR  (WS_COEF  + BATCH*NGROUPP*64)    // B*F*18 face attributes

// ---------------------------------------------------------------- K1: vertices
__global__ void k_vertex(const float* __restrict__ verts,
                         const float* __restrict__ rot,
                         const float* __restrict__ trans,
                         float* __restrict__ ws) {
    int t = blockIdx.x * blockDim.x + threadIdx.x;
    if (t >= BATCH * NV) return;
    int b = t / NV, v = t % NV;
    const float* pv = verts + (b * NV + v) * 3;
    const float* R  = rot   + b * 9;
    const float* T  = trans + b * 3;
    float x = pv[0], y = pv[1], z = pv[2];
    float cx = R[0]*x + R[1]*y + R[2]*z + T[0];
    float cy = R[3]*x + R[4]*y + R[5]*z + T[1];
    float cz = R[6]*x + R[7]*y + R[8]*z + T[2];
    float* vc = ws + WS_VSCAM + (b*NV + v)*3;
    vc[0] = cx; vc[1] = cy; vc[2] = cz;
    float* vi = ws + WS_VSIMG + (b*NV + v)*2;
    vi[0] = cx / cz; vi[1] = cy / cz;
    float* vn = ws + WS_VNORM + (b*NV + v)*3;
    vn[0] = 0.f; vn[1] = 0.f; vn[2] = 0.f;
}

// ---------------------- K2: face normals (scatter) + WMMA-A coefficient setup
// coefficient storage per group (64 floats), lane-order for conflict-free LDS:
//   [0..31]  = {M=0..15} x {k=0(cx), k=1(cy)}   -> float pair at 2*M
//   [32..63] = {M=0..15} x {k=2(c0), k=3(pad)}  -> float pair at 32+2*M
// where M = (f%4)*4 + func, func in {w0,w1,w2,z}
__global__ void k_face(const int* __restrict__ faces, float* __restrict__ ws) {
    int t = blockIdx.x * blockDim.x + threadIdx.x;
    if (t >= BATCH * NFP) return;
    int b = t / NFP, f = t % NFP;
    float* cg = ws + WS_COEF + ((size_t)b*NGROUPP + (f>>2)) * 64;
    int Mb = (f & 3) * 4;

    if (f >= NF) {  // sentinel pad face: always far outside, never wins depth
        #pragma unroll
        for (int func = 0; func < 4; ++func) {
            int M = Mb + func;
            cg[2*M] = 0.f; cg[2*M+1] = 0.f;
            cg[32+2*M] = (func < 3) ? -1e30f : 0.f;
            cg[32+2*M+1] = 0.f;
        }
        return;
    }

    int i0 = faces[f*3+0], i1 = faces[f*3+1], i2 = faces[f*3+2];
    const float* vc = ws + WS_VSCAM + b*NV*3;
    float p0x = vc[i0*3], p0y = vc[i0*3+1], p0z = vc[i0*3+2];
    float p1x = vc[i1*3], p1y = vc[i1*3+1], p1z = vc[i1*3+2];
    float p2x = vc[i2*3], p2y = vc[i2*3+1], p2z = vc[i2*3+2];

    // face normal scatter-add
    float e1x = p1x-p0x, e1y = p1y-p0y, e1z = p1z-p0z;
    float e2x = p2x-p0x, e2y = p2y-p0y, e2z = p2z-p0z;
    float nx = e1y*e2z - e1z*e2y;
    float ny = e1z*e2x - e1x*e2z;
    float nz = e1x*e2y - e1y*e2x;
    float* vn = ws + WS_VNORM + b*NV*3;
    atomicAdd(&vn[i0*3+0], nx); atomicAdd(&vn[i0*3+1], ny); atomicAdd(&vn[i0*3+2], nz);
    atomicAdd(&vn[i1*3+0], nx); atomicAdd(&vn[i1*3+1], ny); atomicAdd(&vn[i1*3+2], nz);
    atomicAdd(&vn[i2*3+0], nx); atomicAdd(&vn[i2*3+1], ny); atomicAdd(&vn[i2*3+2], nz);

    const float* vi = ws + WS_VSIMG + b*NV*2;
    float v0x = vi[i0*2], v0y = vi[i0*2+1];
    float v1x = vi[i1*2], v1y = vi[i1*2+1];
    float v2x = vi[i2*2], v2y = vi[i2*2+1];

    float A = (v1x-v0x)*(v2y-v0y) - (v1y-v0y)*(v2x-v0x);
    if (fabsf(A) < 1e-8f) A = 1e-8f;
    float invA = 1.0f / A;

    // edge(a,b): cx=-(by-ay), cy=(bx-ax), c0=(by-ay)*ax-(bx-ax)*ay ; all /A
    float coef[4][3];
    coef[0][0] = -(v2y - v1y) * invA;
    coef[0][1] =  (v2x - v1x) * invA;
    coef[0][2] = ((v2y - v1y)*v1x - (v2x - v1x)*v1y) * invA;
    coef[1][0] = -(v0y - v2y) * invA;
    coef[1][1] =  (v0x - v2x) * invA;
    coef[1][2] = ((v0y - v2y)*v2x - (v0x - v2x)*v2y) * invA;
    coef[2][0] = -(v1y - v0y) * invA;
    coef[2][1] =  (v1x - v0x) * invA;
    coef[2][2] = ((v1y - v0y)*v0x - (v1x - v0x)*v0y) * invA;
    coef[3][0] = p0z*coef[0][0] + p1z*coef[1][0] + p2z*coef[2][0];
    coef[3][1] = p0z*coef[0][1] + p1z*coef[1][1] + p2z*coef[2][1];
    coef[3][2] = p0z*coef[0][2] + p1z*coef[1][2] + p2z*coef[2][2];

    #pragma unroll
    for (int func = 0; func < 4; ++func) {
        int M = Mb + func;
        cg[2*M]      = coef[func][0];   // cx -> K0
        cg[2*M+1]    = coef[func][1];   // cy -> K1
        cg[32+2*M]   = coef[func][2];   // c0 -> K2
        cg[32+2*M+1] = 0.f;             // pad -> K3
    }
}

// ------------------------------------------------- K3: normalize vertex normals
__global__ void k_vnorm(float* __restrict__ ws) {
    int t = blockIdx.x * blockDim.x + threadIdx.x;
    if (t >= BATCH * NV) return;
    float* vn = ws + WS_VNORM + t*3;
    float x = vn[0], y = vn[1], z = vn[2];
    float inv = 1.0f / (sqrtf(x*x + y*y + z*z) + 1e-10f);
    vn[0] = x*inv; vn[1] = y*inv; vn[2] = z*inv;
}

// ---------------------------- K4: face_attr = concat(corner normals, attribs)
__global__ void k_fattr(const int* __restrict__ faces,
                        const float* __restrict__ attribs,
                        float* __restrict__ ws) {
    int t = blockIdx.x * blockDim.x + threadIdx.x;
    if (t >= BATCH * NF) return;
    int b = t / NF, f = t % NF;
    const float* vn = ws + WS_VNORM + b*NV*3;
    const float* at = attribs + ((size_t)(b*NF + f)) * 9;
    float* fa = ws + WS_FATTR + ((size_t)(b*NF + f)) * 18;
    int idx[3] = { faces[f*3+0], faces[f*3+1], faces[f*3+2] };
    #pragma unroll
    for (int k = 0; k < 3; ++k) {
        int i = idx[k];
        fa[k*6+0] = vn[i*3+0];
        fa[k*6+1] = vn[i*3+1];
        fa[k*6+2] = vn[i*3+2];
        fa[k*6+3] = at[k*3+0];
        fa[k*6+4] = at[k*3+1];
        fa[k*6+5] = at[k*3+2];
    }
}

// ------------------- K5: WMMA rasterizer with async-LDS double-buffer pipeline
// block = 128 threads = 4 waves = 4 pixel tiles (64 pixels) of one batch.
// Per chunk: DMA 16 face-groups (4KB) into LDS with global_load_async_to_lds_b128
// while computing on the previous chunk; per group one v_wmma_f32_16x16x4_f32.
__global__ __launch_bounds__(128) void k_raster(float* __restrict__ out,
                                                const float* __restrict__ ws) {
    __shared__ __align__(16) float sm[2][CHUNK*64];   // 8KB double buffer

    int tid  = threadIdx.x;
    int wave = tid >> 5;
    int l    = tid & 31;
    int n    = l & 15;
    bool hi  = (l >= 16);

    int blk_in_b = blockIdx.x & 63;          // 64 blocks per batch
    int b        = blockIdx.x >> 6;
    int tile     = blk_in_b * 4 + wave;
    int p  = tile * 16 + n;
    int ix = p % IMW, iy = p / IMW;
    float px = -1.0f + 2.0f * (float)ix / (float)(IMW - 1);
    float py = -1.0f + 2.0f * (float)iy / (float)(IMH - 1);

    // B matrix (K4 x N16 pixels): K0=px K1=py K2=1 K3=0
    v2f bm;
    bm.x = hi ? 1.0f : px;
    bm.y = hi ? 0.0f : py;

    const float* cgb = ws + WS_COEF + (size_t)b * NGROUPP * 64;
    const char*  cgbytes = (const char*)cgb;
    unsigned lds0 = (unsigned)(uintptr_t)(void*)&sm[0][0];
    unsigned lds1 = (unsigned)(uintptr_t)(void*)&sm[1][0];

    float zbest  = __builtin_inff();
    int   ibest  = 0x7fffffff;
    float logacc = 0.0f;
    const float onem = 1.0f - 1e-7f;
    int lhalf = hi ? 32 : 0;

    // issue async DMA of one 4KB chunk: 128 threads x 2 x b128 = 4KB
    auto issue = [&](int chunk, unsigned ldsbase) {
        unsigned long long g0 =
            (unsigned long long)(uintptr_t)(cgbytes + (size_t)chunk*CHUNK*256) + (unsigned)tid*16u;
        unsigned long long g1 = g0 + 2048ull;
        unsigned l0 = ldsbase + (unsigned)tid*16u;
        unsigned l1 = l0 + 2048u;
        asm volatile("global_load_async_to_lds_b128 %0, %1, off"
                     :: "v"(l0), "v"(g0) : "memory");
        asm volatile("global_load_async_to_lds_b128 %0, %1, off"
                     :: "v"(l1), "v"(g1) : "memory");
    };

    issue(0, lds0);

    for (int c = 0; c < NCHUNK; ++c) {
        int buf = c & 1;
        if (c + 1 < NCHUNK) {
            issue(c + 1, buf ? lds0 : lds1);
            asm volatile("s_wait_asynccnt 0x2" ::: "memory"); // chunk c landed
        } else {
            asm volatile("s_wait_asynccnt 0x0" ::: "memory");
        }
        __syncthreads();

        const float* smb = &sm[buf][0];
        int fc = c * CHUNK * 4 + (hi ? 2 : 0);
        #pragma unroll 4
        for (int g = 0; g < CHUNK; ++g) {
            v2f a = *(const v2f*)(smb + g*64 + lhalf + n*2);
            v8f acc = {0.f,0.f,0.f,0.f,0.f,0.f,0.f,0.f};
            acc = __builtin_amdgcn_wmma_f32_16x16x4_f32(
                    false, a, false, bm, (short)0, acc, false, false);
            int fidx = fc + g*4;
            #pragma unroll
            for (int j = 0; j < 2; ++j) {
                float w0 = acc[4*j+0], w1 = acc[4*j+1], w2 = acc[4*j+2], zz = acc[4*j+3];
                float dmin = fminf(w0, fminf(w1, w2));   // inside <=> dmin >= 0
                bool take = (dmin >= 0.f) && (zz > 0.f) && (zz < zbest);
                zbest = take ? zz : zbest;
                ibest = take ? (fidx + j) : ibest;
                // prob: exp(clip(dmin/sigma,-30,0)) == 1 when inside (clip->0)
                float t = __expf(fminf(fmaxf(dmin * (1.0f/SIGMA), -30.0f), 0.0f));
                // log1p(-t*onem) via exact-residual fma + log
                logacc += __logf(fmaf(-onem, t, 1.0f));
            }
        }
        __syncthreads();
    }

    // merge lane halves: lane l holds faces %4 in {0,1}; lane l+16 holds {2,3}
    float zo = __shfl_xor(zbest, 16, 32);
    int   io = __shfl_xor(ibest, 16, 32);
    float lo = __shfl_xor(logacc, 16, 32);
    if (zo < zbest || (zo == zbest && io < ibest)) { zbest = zo; ibest = io; }
    logacc += lo;

    if (l < 16) {
        bool covered = zbest < __builtin_inff();
        float at[6] = {0.f,0.f,0.f,0.f,0.f,0.f};
        int idxout = -1;
        if (covered) {
            idxout = ibest;
            const float* gp = cgb + (size_t)(ibest >> 2) * 64;
            int Mb = (ibest & 3) * 4;
            float wb0 = gp[2*(Mb+0)]*px + gp[2*(Mb+0)+1]*py + gp[32+2*(Mb+0)];
            float wb1 = gp[2*(Mb+1)]*px + gp[2*(Mb+1)+1]*py + gp[32+2*(Mb+1)];
            float wb2 = gp[2*(Mb+2)]*px + gp[2*(Mb+2)+1]*py + gp[32+2*(Mb+2)];
            const float* fa = ws + WS_FATTR + ((size_t)(b*NF + ibest)) * 18;
            #pragma unroll
            for (int ch = 0; ch < 6; ++ch)
                at[ch] = wb0*fa[ch] + wb1*fa[6+ch] + wb2*fa[12+ch];
        }
        float nrm = sqrtf(at[0]*at[0] + at[1]*at[1] + at[2]*at[2]);
        float inv = 1.0f / (nrm + 1e-10f);

        int po = b * NPIX + p;
        float* out_n = out;                     // imnormal B*H*W*3
        float* out_a = out + BATCH*NPIX*3;      // im_attr  B*H*W*3
        float* out_p = out + BATCH*NPIX*6;      // improb   B*H*W
        float* out_i = out + BATCH*NPIX*7;      // imdx     B*H*W
        out_n[po*3+0] = at[0]*inv;
        out_n[po*3+1] = at[1]*inv;
        out_n[po*3+2] = at[2]*inv;
        out_a[po*3+0] = at[3];
        out_a[po*3+1] = at[4];
        out_a[po*3+2] = at[5];
        out_p[po] = 1.0f - __expf(logacc);
        out_i[po] = (float)idxout;
    }
}

extern "C" void kernel_launch(void* const* d_in, const int* in_sizes, int n_in,
                              void* d_out, int out_size, void* d_ws, size_t ws_size,
                              hipStream_t stream) {
    const float* verts   = (const float*)d_in[0];
    const int*   faces   = (const int*)  d_in[1];
    const float* attribs = (const float*)d_in[2];
    const float* rot     = (const float*)d_in[3];
    const float* trans   = (const float*)d_in[4];
    float* ws  = (float*)d_ws;
    float* out = (float*)d_out;

    k_vertex<<<(BATCH*NV  + 255)/256, 256, 0, stream>>>(verts, rot, trans, ws);
    k_face  <<<(BATCH*NFP + 255)/256, 256, 0, stream>>>(faces, ws);
    k_vnorm <<<(BATCH*NV  + 255)/256, 256, 0, stream>>>(ws);
    k_fattr <<<(BATCH*NF  + 255)/256, 256, 0, stream>>>(faces, attribs, ws);
    k_raster<<<BATCH*(NPIX/64), 128, 0, stream>>>(out, ws);
}